// qcir_601295421979
// MI455X (gfx1250) — compile-verified
//
#include <hip/hip_runtime.h>
#include <math.h>

#define NQ 13
#define DIM 8192            // 2^13
#define REPEAT 6
#define NLAYER 7            // 6 entangling + 1 final YZY
#define THREADS 256         // 8 wave32
#define SPT (DIM / THREADS) // 32 states per thread

typedef __attribute__((ext_vector_type(2))) float v2f;
typedef __attribute__((ext_vector_type(8))) float v8f;

// CNOT ring as basis permutation: v_out[i] = v_in[perm(i)]
__device__ __forceinline__ int ring_perm(int m) {
#pragma unroll
  for (int c = NQ - 1; c >= 0; --c) {
    int t = (c + 1) % NQ;
    int bit = (m >> (NQ - 1 - c)) & 1;
    m ^= bit << (NQ - 1 - t);
  }
  return m;
}

// State index from (group G, group-row r in [0,16), group-col c in [0,512)).
// Qubit 0 is the MSB of the 13-bit index. G0 = qubits 0-3 (bits 12..9),
// G1 = qubits 4-7 (bits 8..5), G2 = qubits 8-11 (bits 4..1); qubit 12 = bit 0.
template <int G>
__device__ __forceinline__ int gidx(int r, int c) {
  if constexpr (G == 0) {
    return (r << 9) | c;
  } else if constexpr (G == 1) {
    return ((c >> 5) << 9) | (r << 5) | (c & 31);
  } else {
    return ((c >> 1) << 5) | (r << 1) | (c & 1);
  }
}

// Build fused 16x16 unitary for (layer L, group g): one entry per thread.
__device__ __forceinline__ void build_U(int L, int g, const float* gate,
                                        float* Ure, float* Uim, int tid) {
  int r = tid >> 4, c = tid & 15;
  float re = 1.f, im = 0.f;
#pragma unroll
  for (int j = 0; j < 4; ++j) {
    const float* gp = gate + (L * NQ + g * 4 + j) * 8;
    int rb = (r >> (3 - j)) & 1, cb2 = (c >> (3 - j)) & 1;
    float er = gp[(rb * 2 + cb2) * 2], ei = gp[(rb * 2 + cb2) * 2 + 1];
    float nr = re * er - im * ei, ni = re * ei + im * er;
    re = nr; im = ni;
  }
  Ure[r * 17 + c] = re;
  Uim[r * 17 + c] = im;
}

// Apply the fused 16x16 complex unitary to the state for group G.
// FQ12: additionally apply the qubit-12 2x2 gate to the accumulators via
//       adjacent-lane exchange (valid for G==2, where qubit 12 == c&1 == lm&1).
// LAST: instead of scattering, accumulate signed |amp|^2 partials into zl[NQ].
// Each wave owns 4 disjoint 16-column tiles -> no barriers inside the pass.
template <int G, bool FQ12, bool LAST>
__device__ __forceinline__ void group_pass(const float* Ure, const float* Uim,
                                           float* sRe, float* sIm,
                                           const float* gp12, float* zl,
                                           int wv, int hi, int lm) {
  // A fragments (16x4 f32 layout: M = lane&15; VGPR v holds K = 4k + v + 2*hi)
  v2f Ar[4], Ai[4], An[4];
#pragma unroll
  for (int k = 0; k < 4; ++k) {
#pragma unroll
    for (int v = 0; v < 2; ++v) {
      int K = 4 * k + v + 2 * hi;
      float re = Ure[lm * 17 + K];
      float im = Uim[lm * 17 + K];
      Ar[k][v] = re; Ai[k][v] = im; An[k][v] = -im;
    }
  }
  // qubit-12 coefficients, parity-selected per lane:
  // even lane (c&1==0): new = u00*own + u01*other; odd: new = u11*own + u10*other
  float cAr = 0.f, cAi = 0.f, cBr = 0.f, cBi = 0.f;
  if constexpr (FQ12) {
    bool even = ((lm & 1) == 0);
    cAr = even ? gp12[0] : gp12[6];
    cAi = even ? gp12[1] : gp12[7];
    cBr = even ? gp12[2] : gp12[4];
    cBi = even ? gp12[3] : gp12[5];
  }
#pragma unroll
  for (int t4 = 0; t4 < 4; ++t4) {
    const int col = (wv * 4 + t4) * 16 + lm;
    v2f Br[4], Bi[4];
#pragma unroll
    for (int k = 0; k < 4; ++k) {
#pragma unroll
      for (int v = 0; v < 2; ++v) {
        int K = 4 * k + v + 2 * hi;
        int a = gidx<G>(K, col);
        Br[k][v] = sRe[a]; Bi[k][v] = sIm[a];
      }
    }
    v8f Dr = {0, 0, 0, 0, 0, 0, 0, 0};
    v8f Di = {0, 0, 0, 0, 0, 0, 0, 0};
    // Re = Ur*Br - Ui*Bi (minus folded into An); Im = Ur*Bi + Ui*Br
#pragma unroll
    for (int k = 0; k < 4; ++k)
      Dr = __builtin_amdgcn_wmma_f32_16x16x4_f32(false, Ar[k], false, Br[k], (short)0, Dr, false, false);
#pragma unroll
    for (int k = 0; k < 4; ++k)
      Dr = __builtin_amdgcn_wmma_f32_16x16x4_f32(false, An[k], false, Bi[k], (short)0, Dr, false, false);
#pragma unroll
    for (int k = 0; k < 4; ++k)
      Di = __builtin_amdgcn_wmma_f32_16x16x4_f32(false, Ar[k], false, Bi[k], (short)0, Di, false, false);
#pragma unroll
    for (int k = 0; k < 4; ++k)
      Di = __builtin_amdgcn_wmma_f32_16x16x4_f32(false, Ai[k], false, Br[k], (short)0, Di, false, false);

    if constexpr (FQ12) {
#pragma unroll
      for (int j = 0; j < 8; ++j) {
        float orr = __shfl_xor((float)Dr[j], 1, 32);
        float oii = __shfl_xor((float)Di[j], 1, 32);
        float nr = cAr * Dr[j] - cAi * Di[j] + cBr * orr - cBi * oii;
        float ni = cAr * Di[j] + cAi * Dr[j] + cBr * oii + cBi * orr;
        Dr[j] = nr; Di[j] = ni;
      }
    }
    if constexpr (LAST) {
      // accumulate <Z_q> partials directly from final amplitudes
#pragma unroll
      for (int j = 0; j < 8; ++j) {
        int row = j + 8 * hi;
        int s = gidx<G>(row, col);
        float pr = Dr[j] * Dr[j] + Di[j] * Di[j];
#pragma unroll
        for (int q = 0; q < NQ; ++q)
          zl[q] += ((s >> (NQ - 1 - q)) & 1) ? -pr : pr;
      }
    } else {
      // scatter D (C/D layout: VGPR j holds row j + 8*hi, col = lane&15)
#pragma unroll
      for (int j = 0; j < 8; ++j) {
        int row = j + 8 * hi;
        int a = gidx<G>(row, col);
        sRe[a] = Dr[j]; sIm[a] = Di[j];
      }
    }
  }
}

__global__ __launch_bounds__(THREADS) void qsim13_kernel(
    const float* __restrict__ x,   // [B, NQ]
    const float* __restrict__ w,   // [21*NQ]
    float* __restrict__ out)       // [B, NQ]
{
  extern __shared__ float smem[];
  float* sRe  = smem;                       // DIM
  float* sIm  = sRe + DIM;                  // DIM
  float* gate = sIm + DIM;                  // NLAYER*NQ*8 (2x2 complex gates)
  float* Ure  = gate + NLAYER * NQ * 8;     // 16*17 (padded fused unitary, real)
  float* Uim  = Ure + 16 * 17;              // 16*17 (imag)
  float* vtab = Uim + 16 * 17;              // NQ*4 (encoding vectors)
  float* wacc = vtab + NQ * 4;              // 8*NQ (per-wave reduction partials)

  const int tid  = threadIdx.x;
  const int lane = tid & 31;
  const int wv   = tid >> 5;
  const int hi   = lane >> 4;   // 0: lanes 0-15, 1: lanes 16-31
  const int lm   = lane & 15;
  const int b    = blockIdx.x;

  // ---- per-qubit Pauli-encoding vectors: v0 = cos(h)e^{-ih}, v1 = sin(h)e^{ih} ----
  if (tid < NQ) {
    float h = 0.5f * x[b * NQ + tid];
    float sh, ch; sincosf(h, &sh, &ch);
    vtab[tid * 4 + 0] = ch * ch;  vtab[tid * 4 + 1] = -ch * sh;
    vtab[tid * 4 + 2] = sh * ch;  vtab[tid * 4 + 3] =  sh * sh;
  }
  // ---- 2x2 gates: u = Ry(y2) * Rz(z) * Ry(y1), one thread per (layer,qubit) ----
  if (tid < NLAYER * NQ) {
    int L = tid / NQ, q = tid % NQ;
    const float* p = w + 3 * NQ * L + 3 * q;   // final layer sits at L==6
    float sa, ca; sincosf(0.5f * p[0], &sa, &ca);
    float sz, cz; sincosf(0.5f * p[1], &sz, &cz);
    float sb, cb; sincosf(0.5f * p[2], &sb, &cb);
    // M = Rz(z)*Ry(a): row0 = (cz - i sz)[ca, -sa], row1 = (cz + i sz)[sa, ca]
    float m00r =  ca * cz, m00i = -ca * sz;
    float m01r = -sa * cz, m01i =  sa * sz;
    float m10r =  sa * cz, m10i =  sa * sz;
    float m11r =  ca * cz, m11i =  ca * sz;
    float* gp = gate + tid * 8;
    gp[0] = cb * m00r - sb * m10r;  gp[1] = cb * m00i - sb * m10i; // u00
    gp[2] = cb * m01r - sb * m11r;  gp[3] = cb * m01i - sb * m11i; // u01
    gp[4] = sb * m00r + cb * m10r;  gp[5] = sb * m00i + cb * m10i; // u10
    gp[6] = sb * m01r + cb * m11r;  gp[7] = sb * m01i + cb * m11i; // u11
  }
  __syncthreads();

  // ---- initial product state ----
#pragma unroll
  for (int i = 0; i < SPT; ++i) {
    int s = tid + THREADS * i;
    float re = 1.f, im = 0.f;
#pragma unroll
    for (int q = 0; q < NQ; ++q) {
      int bit = (s >> (NQ - 1 - q)) & 1;
      float vr = vtab[q * 4 + 2 * bit], vi = vtab[q * 4 + 2 * bit + 1];
      float nr = re * vr - im * vi, ni = re * vi + im * vr;
      re = nr; im = ni;
    }
    sRe[s] = re; sIm[s] = im;
  }

  float zl[NQ];
#pragma unroll
  for (int q = 0; q < NQ; ++q) zl[q] = 0.f;

  // ---- 6 entangling layers: 3 WMMA group passes (q12 fused into group 2),
  //      then the CNOT-ring permutation ----
  for (int L = 0; L < REPEAT; ++L) {
    const float* gp12 = gate + (L * NQ + 12) * 8;

    __syncthreads();
    build_U(L, 0, gate, Ure, Uim, tid);
    __syncthreads();
    group_pass<0, false, false>(Ure, Uim, sRe, sIm, gp12, zl, wv, hi, lm);

    __syncthreads();
    build_U(L, 1, gate, Ure, Uim, tid);
    __syncthreads();
    group_pass<1, false, false>(Ure, Uim, sRe, sIm, gp12, zl, wv, hi, lm);

    __syncthreads();
    build_U(L, 2, gate, Ure, Uim, tid);
    __syncthreads();
    group_pass<2, true, false>(Ure, Uim, sRe, sIm, gp12, zl, wv, hi, lm);

    // CNOT-ring permutation (in place via register staging)
    __syncthreads();
    {
      float tr[SPT], ti[SPT];
#pragma unroll
      for (int i = 0; i < SPT; ++i) {
        int s = tid + THREADS * i;
        int src = ring_perm(s);
        tr[i] = sRe[src]; ti[i] = sIm[src];
      }
      __syncthreads();
#pragma unroll
      for (int i = 0; i < SPT; ++i) {
        int s = tid + THREADS * i;
        sRe[s] = tr[i]; sIm[s] = ti[i];
      }
    }
  }

  // ---- final YZY layer; expectation accumulated straight from the
  //      group-2 accumulators (no scatter, no state re-read) ----
  {
    const float* gp12 = gate + (REPEAT * NQ + 12) * 8;

    __syncthreads();
    build_U(REPEAT, 0, gate, Ure, Uim, tid);
    __syncthreads();
    group_pass<0, false, false>(Ure, Uim, sRe, sIm, gp12, zl, wv, hi, lm);

    __syncthreads();
    build_U(REPEAT, 1, gate, Ure, Uim, tid);
    __syncthreads();
    group_pass<1, false, false>(Ure, Uim, sRe, sIm, gp12, zl, wv, hi, lm);

    __syncthreads();
    build_U(REPEAT, 2, gate, Ure, Uim, tid);
    __syncthreads();
    group_pass<2, true, true>(Ure, Uim, sRe, sIm, gp12, zl, wv, hi, lm);
  }

  // ---- deterministic shuffle-tree + per-wave partials ----
#pragma unroll
  for (int q = 0; q < NQ; ++q) {
    float v = zl[q];
    for (int off = 16; off > 0; off >>= 1) v += __shfl_down(v, off, 32);
    if (lane == 0) wacc[wv * NQ + q] = v;
  }
  __syncthreads();
  if (tid < NQ) {
    float acc = 0.f;
#pragma unroll
    for (int k = 0; k < THREADS / 32; ++k) acc += wacc[k * NQ + tid];
    out[b * NQ + tid] = 4.0f * acc;
  }
}

extern "C" void kernel_launch(void* const* d_in, const int* in_sizes, int n_in,
                              void* d_out, int out_size, void* d_ws, size_t ws_size,
                              hipStream_t stream) {
  const float* x  = (const float*)d_in[0];   // [B, NQ] f32
  const float* wt = (const float*)d_in[1];   // [21*NQ] f32
  float* out = (float*)d_out;                // [B, NQ] f32
  int B = in_sizes[0] / NQ;                  // 2048
  size_t shmem = (size_t)(2 * DIM + NLAYER * NQ * 8 + 2 * 16 * 17 + NQ * 4 + 8 * NQ)
                 * sizeof(float);            // ~71 KB, fits 320 KB/WGP LDS
  qsim13_kernel<<<B, THREADS, shmem, stream>>>(x, wt, out);
}